// PortfolioGenerator_5188320493903
// MI455X (gfx1250) — compile-verified
//
#include <hip/hip_runtime.h>
#include <stdint.h>

#define I_DIM   8192
#define G_WIN   7
#define THREADS 1024
#define EPT     (I_DIM / THREADS)          // 8 elements per thread
#define LDS_SLOTS (I_DIM + I_DIM / 8)      // 9216 u64 = 72 KB (padded)

typedef __attribute__((address_space(3))) void lds_void_t;

__device__ __forceinline__ unsigned lds_byte_off(void* p) {
    return (unsigned)(uintptr_t)(lds_void_t*)p;   // AS(3) ptr value == LDS byte offset
}

// bank-conflict-avoiding slot: one pad slot per 8 elements
__device__ __forceinline__ unsigned sl(unsigned e) { return e + (e >> 3); }

// monotone f32 <-> u32 (order-preserving)
__device__ __forceinline__ unsigned f32_mono(unsigned u) {
    return (u & 0x80000000u) ? ~u : (u | 0x80000000u);
}
__device__ __forceinline__ float mono_to_f32(unsigned m) {
    unsigned u = (m & 0x80000000u) ? (m & 0x7fffffffu) : ~m;
    return __uint_as_float(u);
}

// dir=true -> keep larger first (descending)
__device__ __forceinline__ void cmpswap(unsigned long long& x, unsigned long long& y, bool dir) {
    if (dir ? (x < y) : (x > y)) { unsigned long long t = x; x = y; y = t; }
}

// 3 bitonic levels (strides 4,2,1 in the local array) with uniform direction
__device__ __forceinline__ void sort8_stage(unsigned long long a[8], bool dir) {
    #pragma unroll
    for (int i = 0; i < 4; ++i) cmpswap(a[i], a[i + 4], dir);
    #pragma unroll
    for (int h = 0; h < 8; h += 4) {
        cmpswap(a[h + 0], a[h + 2], dir);
        cmpswap(a[h + 1], a[h + 3], dir);
    }
    #pragma unroll
    for (int h = 0; h < 8; h += 2) cmpswap(a[h], a[h + 1], dir);
}

// 2 bitonic levels (strides 2,1 in the local array) with uniform direction
__device__ __forceinline__ void sort4_stage(unsigned long long a[4], bool dir) {
    cmpswap(a[0], a[2], dir); cmpswap(a[1], a[3], dir);
    cmpswap(a[0], a[1], dir); cmpswap(a[2], a[3], dir);
}

// stages k=2,4,8 entirely in registers; dir8 = direction of the k=8 stage
__device__ __forceinline__ void local_presort(unsigned long long a[8], bool dir8) {
    cmpswap(a[0], a[1], true);  cmpswap(a[2], a[3], false);
    cmpswap(a[4], a[5], true);  cmpswap(a[6], a[7], false);
    cmpswap(a[0], a[2], true);  cmpswap(a[1], a[3], true);
    cmpswap(a[4], a[6], false); cmpswap(a[5], a[7], false);
    cmpswap(a[0], a[1], true);  cmpswap(a[2], a[3], true);
    cmpswap(a[4], a[5], false); cmpswap(a[6], a[7], false);
    sort8_stage(a, dir8);
}

// one plain pairwise pass at level j of stage k (4 pairs/thread)
__device__ __forceinline__ void pass1(unsigned long long* keys, unsigned tid,
                                      unsigned j, unsigned k) {
    #pragma unroll
    for (int p = 0; p < EPT / 2; ++p) {
        unsigned i  = tid + (unsigned)p * THREADS;
        unsigned lo = ((i & ~(j - 1u)) << 1) | (i & (j - 1u));
        unsigned hi = lo | j;
        unsigned long long a = keys[sl(lo)];
        unsigned long long b = keys[sl(hi)];
        bool up = (lo & k) == 0;
        if (up ? (a < b) : (a > b)) { keys[sl(lo)] = b; keys[sl(hi)] = a; }
    }
}

// levels jj, jj/2 of stage k as 4-element strided register groups (2 groups/thread)
__device__ __forceinline__ void round4(unsigned long long* keys, unsigned tid,
                                       unsigned jj, unsigned k) {
    unsigned s = jj >> 1;
    #pragma unroll
    for (int p = 0; p < 2; ++p) {
        unsigned g = tid + (unsigned)p * THREADS;
        unsigned b = ((g & ~(s - 1u)) << 2) | (g & (s - 1u));
        unsigned long long a[4];
        #pragma unroll
        for (int m = 0; m < 4; ++m) a[m] = keys[sl(b + (unsigned)m * s)];
        sort4_stage(a, (b & k) == 0);
        #pragma unroll
        for (int m = 0; m < 4; ++m) keys[sl(b + (unsigned)m * s)] = a[m];
    }
}

// levels jj, jj/2, jj/4 of stage k as 8-element strided register groups (1 group/thread)
__device__ __forceinline__ void round8(unsigned long long* keys, unsigned tid,
                                       unsigned jj, unsigned k) {
    unsigned s = jj >> 2;
    unsigned b = ((tid & ~(s - 1u)) << 3) | (tid & (s - 1u));
    unsigned long long a[8];
    #pragma unroll
    for (int m = 0; m < 8; ++m) a[m] = keys[sl(b + (unsigned)m * s)];
    sort8_stage(a, (b & k) == 0);
    #pragma unroll
    for (int m = 0; m < 8; ++m) keys[sl(b + (unsigned)m * s)] = a[m];
}

__global__ __launch_bounds__(THREADS, 1)
void portfolio_sort_kernel(const float* __restrict__ x,
                           float* __restrict__ out_bc,
                           float* __restrict__ out_idx) {
    __shared__ unsigned long long keys[LDS_SLOTS];

    const int row = blockIdx.x;
    const unsigned tid = threadIdx.x;
    const float* xrow  = x       + (size_t)row * I_DIM;
    float*       bcrow = out_bc  + (size_t)row * I_DIM;
    float*       ixrow = out_idx + (size_t)row * I_DIM;

    const unsigned kbase = lds_byte_off(keys);

    // ---- 1) async global -> LDS: scatter x[e] into HIGH word of keys[sl(e)]
    #pragma unroll
    for (int p = 0; p < EPT; ++p) {
        unsigned e  = tid + (unsigned)p * THREADS;             // coalesced global
        unsigned la = kbase + sl(e) * 8u + 4u;                 // little-endian high half
        unsigned long long ga = (unsigned long long)(uintptr_t)(xrow + e);
        asm volatile("global_load_async_to_lds_b32 %0, %1, off"
                     :: "v"(la), "v"(ga) : "memory");
    }

    // ---- 2) zero-fill b_c middle while async copy is in flight
    #pragma unroll
    for (int p = 0; p < EPT; ++p) {
        unsigned e = tid + (unsigned)p * THREADS;
        if (e >= G_WIN && e < I_DIM - G_WIN) bcrow[e] = 0.0f;
    }

#if __has_builtin(__builtin_amdgcn_s_wait_asynccnt)
    __builtin_amdgcn_s_wait_asynccnt(0);
#else
    asm volatile("s_wait_asynccnt 0x0" ::: "memory");
#endif
    __syncthreads();

    // ---- 3) build keys + stages k=2..8 fully in registers (one LDS round-trip)
    {
        const unsigned base = tid * 8u;
        unsigned long long a[8];
        #pragma unroll
        for (int i = 0; i < 8; ++i) {
            unsigned e = base + (unsigned)i;
            unsigned m = f32_mono((unsigned)(keys[sl(e)] >> 32));
            a[i] = ((unsigned long long)m << 32) | (unsigned)(I_DIM - 1 - e);
        }
        local_presort(a, (tid & 1u) == 0);       // dir of k=8 stage: (8t & 8)==0
        #pragma unroll
        for (int i = 0; i < 8; ++i) keys[sl(base + (unsigned)i)] = a[i];
    }

    // ---- 4) stages k=16..8192: levels processed top-down in chunks of 3/2/1.
    //         Keep rolled: the schedule is wave-uniform SALU control; a small
    //         resident loop body beats 40 KB of unrolled straight-line code.
    #pragma unroll 1
    for (unsigned k = 16, m = 4; k <= I_DIM; k <<= 1, ++m) {
        unsigned j = k >> 1;
        unsigned r = m % 3;
        if (r == 1) {                 // single level j
            __syncthreads();
            pass1(keys, tid, j, k);
            j >>= 1;
        } else if (r == 2) {          // levels j, j/2
            __syncthreads();
            round4(keys, tid, j, k);
            j >>= 2;
        }
        #pragma unroll 1
        for (; j >= 4; j >>= 3) {     // levels j, j/2, j/4
            __syncthreads();
            round8(keys, tid, j, k);
        }
    }
    __syncthreads();

    // ---- 5) sorted indices (exact in f32 since < 2^24), coalesced
    #pragma unroll
    for (int p = 0; p < EPT; ++p) {
        unsigned e   = tid + (unsigned)p * THREADS;
        unsigned idx = (I_DIM - 1u) - (unsigned)(keys[sl(e)] & 0xffffffffu);
        ixrow[e] = (float)idx;
    }

    // ---- 6) winner softmax (thread 0), loser softmax (thread 32, separate wave)
    if (tid == 0) {
        float v[G_WIN], ex[G_WIN], mx = -3.402823466e38f, s = 0.f;
        #pragma unroll
        for (int i = 0; i < G_WIN; ++i) {
            v[i] = mono_to_f32((unsigned)(keys[sl((unsigned)i)] >> 32));
            mx = fmaxf(mx, v[i]);
        }
        #pragma unroll
        for (int i = 0; i < G_WIN; ++i) { ex[i] = __expf(v[i] - mx); s += ex[i]; }
        #pragma unroll
        for (int i = 0; i < G_WIN; ++i) bcrow[i] = ex[i] / s;
    }
    if (tid == 32) {
        float v[G_WIN], ex[G_WIN], mx = -3.402823466e38f, s = 0.f;
        #pragma unroll
        for (int i = 0; i < G_WIN; ++i) {
            v[i] = 1.0f - mono_to_f32((unsigned)(keys[sl((unsigned)(I_DIM - G_WIN + i))] >> 32));
            mx = fmaxf(mx, v[i]);
        }
        #pragma unroll
        for (int i = 0; i < G_WIN; ++i) { ex[i] = __expf(v[i] - mx); s += ex[i]; }
        #pragma unroll
        for (int i = 0; i < G_WIN; ++i) bcrow[I_DIM - G_WIN + i] = -ex[i] / s;
    }
}

extern "C" void kernel_launch(void* const* d_in, const int* in_sizes, int n_in,
                              void* d_out, int out_size, void* d_ws, size_t ws_size,
                              hipStream_t stream) {
    (void)n_in; (void)d_ws; (void)ws_size; (void)out_size;
    const float* x = (const float*)d_in[0];
    float* out = (float*)d_out;
    int B = in_sizes[0] / I_DIM;                 // 4096
    float* out_bc  = out;
    float* out_idx = out + (size_t)B * I_DIM;    // outputs concatenated flat
    portfolio_sort_kernel<<<B, THREADS, 0, stream>>>(x, out_bc, out_idx);
}